// CapsuleLayer_blk_pru_24919400251517
// MI455X (gfx1250) — compile-verified
//
#include <hip/hip_runtime.h>
#include <hip/hip_bf16.h>
#include <math.h>

// CapsuleNet dynamic routing on MI455X (gfx1250, wave32).
// B=256, I=8 (in capsule dim), C=1152 (in capsules), U=10 (out capsules), J=16 (out dim).

#define B_SZ   256
#define C_CH   1152
#define NUM_U  10
#define J_SZ   16
#define I_SZ   8
#define UJ     160                       // NUM_U * J_SZ
#define BI     2048                      // B_SZ * I_SZ
#define UHAT_PER_C 40960                 // B_SZ * UJ
#define BU     11520                     // C_CH * NUM_U

typedef __attribute__((ext_vector_type(2))) float v2f;
typedef __attribute__((ext_vector_type(4))) float v4f;
typedef __attribute__((ext_vector_type(8))) float v8f;

// ---------------------------------------------------------------------------
// Kernel 0: transpose x [BI=2048 rows, C=1152 cols] -> xT [C, BI]
// so each channel's A-operand (256x8) is contiguous for the WMMA kernel.
// ---------------------------------------------------------------------------
__global__ __launch_bounds__(256) void caps_transpose(const float* __restrict__ x,
                                                      float* __restrict__ xT) {
    __shared__ float tile[32][33];
    const int c0 = blockIdx.x * 32;      // channel tile (36 tiles)
    const int r0 = blockIdx.y * 32;      // (b*8+i) tile  (64 tiles)
    const int tx = threadIdx.x;          // 0..31
    const int ty = threadIdx.y;          // 0..7
#pragma unroll
    for (int k = 0; k < 4; ++k) {
        const int r = r0 + ty + k * 8;
        tile[ty + k * 8][tx] = x[(size_t)r * C_CH + c0 + tx];
    }
    __syncthreads();
#pragma unroll
    for (int k = 0; k < 4; ++k) {
        const int cc = ty + k * 8;
        xT[(size_t)(c0 + cc) * BI + r0 + tx] = tile[tx][cc];
    }
}

// ---------------------------------------------------------------------------
// Kernel 1: u_hat[c][b][u*16+j] = sum_i xT[c][b*8+i] * W[c][u][j][i]
// Per channel: GEMM M=256 (b), N=160 (u,j), K=8 (i) using V_WMMA_F32_16X16X4_F32.
// 1152 blocks x 320 threads (10 waves). Wave w owns N-tile u=w (B operand loaded
// once, reused over 16 M-tiles); K=8 -> two chained WMMAs per tile.
// A 16x4 f32 layout: lanes0-15 {K0,K1}, lanes16-31 {K2,K3}; B mirrored.
// Block 0 also initializes b_ij = 0 and c_ij = softmax(0) = 0.1.
// ---------------------------------------------------------------------------
__global__ __launch_bounds__(320) void caps_uhat(const float* __restrict__ xT,
                                                 const float* __restrict__ W,
                                                 float* __restrict__ uhat,
                                                 float* __restrict__ bij,
                                                 float* __restrict__ cij) {
    const int c = blockIdx.x;
    __shared__ float Xc[BI];             // 8 KB: this channel's 256x8 A matrix

    // Stage A matrix into LDS, coalesced float4 loads.
    {
        const v4f* src = (const v4f*)(xT + (size_t)c * BI);
        v4f* dst = (v4f*)Xc;
        for (int idx = threadIdx.x; idx < BI / 4; idx += 320) dst[idx] = src[idx];
    }
    if (blockIdx.x == 0) {               // init routing state once
        for (int i = threadIdx.x; i < BU; i += 320) { bij[i] = 0.0f; cij[i] = 0.1f; }
    }
    __syncthreads();

    const int wave = threadIdx.x >> 5;   // 0..9  == u
    const int lane = threadIdx.x & 31;
    const int lm   = lane & 15;          // row/col within tile
    const int hi   = lane >> 4;          // half-wave select
    const int kb   = hi * 2;             // K base for this half: 0 or 2

    // B operand (4x16 slice of W^T), constant across all M-tiles of this wave.
    const float* Wb = W + (((size_t)c * NUM_U + wave) * J_SZ + lm) * I_SZ;
    v2f b0, b1;
    b0.x = Wb[kb];     b0.y = Wb[kb + 1];        // K = kb, kb+1   (K 0..3)
    b1.x = Wb[4 + kb]; b1.y = Wb[4 + kb + 1];    // K = 4+kb, ...  (K 4..7)

    float* outBase = uhat + (size_t)c * UHAT_PER_C + wave * J_SZ + lm;

#pragma unroll 4
    for (int mt = 0; mt < 16; ++mt) {
        const float* Xa = Xc + (mt * 16 + lm) * I_SZ + kb;
        v2f a0; a0.x = Xa[0]; a0.y = Xa[1];
        v2f a1; a1.x = Xa[4]; a1.y = Xa[5];
        v8f acc = {};
        acc = __builtin_amdgcn_wmma_f32_16x16x4_f32(false, a0, false, b0,
                                                    (short)0, acc, false, false);
        acc = __builtin_amdgcn_wmma_f32_16x16x4_f32(false, a1, false, b1,
                                                    (short)0, acc, false, false);
        // D layout: VGPR r holds row (r + 8*hi), col = lm.
        float* o = outBase + (size_t)(mt * 16 + hi * 8) * UJ;
#pragma unroll
        for (int r = 0; r < 8; ++r) o[(size_t)r * UJ] = acc[r];
    }
}

// ---------------------------------------------------------------------------
// Kernel 2: s[b,u,j] = sum_c c_ij[c,u] * u_hat[c][b][uj]; v = squash(s).
// 160 blocks x 256 threads; thread g owns one (b,u,j); streams 188.7 MB over C
// with perfectly coalesced reads (all lanes hit one contiguous channel slab).
// Squash's |s|^2 over the 16 j's = shfl_xor butterfly inside each half-wave.
// ---------------------------------------------------------------------------
__global__ __launch_bounds__(256) void caps_route_sv(const float* __restrict__ uhat,
                                                     const float* __restrict__ cij,
                                                     float* __restrict__ vout) {
    __shared__ float cs[BU];             // 46 KB staged coupling coefficients
    for (int i = threadIdx.x; i < BU; i += 256) cs[i] = cij[i];
    __syncthreads();

    const int g  = blockIdx.x * 256 + threadIdx.x;   // 0..40959
    const int u  = (g % UJ) >> 4;
    const float* up = uhat + g;

    float acc = 0.0f;
#pragma unroll 1
    for (int c = 0; c < C_CH; c += 4) {
        __builtin_prefetch(up + (size_t)(c + 16) * UHAT_PER_C, 0, 0);
        const float a0 = up[(size_t)(c + 0) * UHAT_PER_C];
        const float a1 = up[(size_t)(c + 1) * UHAT_PER_C];
        const float a2 = up[(size_t)(c + 2) * UHAT_PER_C];
        const float a3 = up[(size_t)(c + 3) * UHAT_PER_C];
        acc = fmaf(a0, cs[(c + 0) * NUM_U + u], acc);
        acc = fmaf(a1, cs[(c + 1) * NUM_U + u], acc);
        acc = fmaf(a2, cs[(c + 2) * NUM_U + u], acc);
        acc = fmaf(a3, cs[(c + 3) * NUM_U + u], acc);
    }

    // |s|^2 over j (16 consecutive lanes, aligned inside the wave32 halves)
    float ss = acc * acc;
    ss += __shfl_xor(ss, 1, 32);
    ss += __shfl_xor(ss, 2, 32);
    ss += __shfl_xor(ss, 4, 32);
    ss += __shfl_xor(ss, 8, 32);

    const float scale = ss / ((1.0f + ss) * sqrtf(ss));
    vout[g] = acc * scale;
}

// ---------------------------------------------------------------------------
// Kernel 3: u_vj[c,u] = (1/B) sum_{b,j} u_hat[c][b][uj]*v[b][uj];
//           b_ij += u_vj; c_ij = softmax_u(b_ij).  One block per channel.
// 320 threads: t -> (bsub = t/160, uj = t%160); coalesced sweep over b.
// ---------------------------------------------------------------------------
__global__ __launch_bounds__(320) void caps_agree(const float* __restrict__ uhat,
                                                  const float* __restrict__ v,
                                                  float* __restrict__ bij,
                                                  float* __restrict__ cij) {
    const int c  = blockIdx.x;
    const int t  = threadIdx.x;
    const int uj = t % UJ;
    const int bs = t / UJ;               // 0 or 1
    const float* uc = uhat + (size_t)c * UHAT_PER_C;

    float acc = 0.0f;
#pragma unroll 4
    for (int b = bs; b < B_SZ; b += 2) {
        const int off = b * UJ + uj;
        acc = fmaf(uc[off], v[off], acc);
    }
    // reduce over j (16 lanes)
    acc += __shfl_xor(acc, 1, 32);
    acc += __shfl_xor(acc, 2, 32);
    acc += __shfl_xor(acc, 4, 32);
    acc += __shfl_xor(acc, 8, 32);

    __shared__ float part[32];
    if ((t & 15) == 0) part[bs * NUM_U + (uj >> 4)] = acc;
    __syncthreads();

    if (t < NUM_U) {
        const float uv = (part[t] + part[NUM_U + t]) * (1.0f / (float)B_SZ);
        const float bn = bij[c * NUM_U + t] + uv;
        bij[c * NUM_U + t] = bn;
        part[t] = bn;                    // reuse for softmax
    }
    __syncthreads();
    if (t < NUM_U) {
        float m = part[0];
#pragma unroll
        for (int i = 1; i < NUM_U; ++i) m = fmaxf(m, part[i]);
        float sum = 0.0f;
#pragma unroll
        for (int i = 0; i < NUM_U; ++i) sum += expf(part[i] - m);
        cij[c * NUM_U + t] = expf(part[t] - m) / sum;
    }
}

// ---------------------------------------------------------------------------
extern "C" void kernel_launch(void* const* d_in, const int* in_sizes, int n_in,
                              void* d_out, int out_size, void* d_ws, size_t ws_size,
                              hipStream_t stream) {
    const float* x = (const float*)d_in[0];   // [B, I, C]
    const float* W = (const float*)d_in[1];   // [1, C, U, J, I]
    (void)in_sizes; (void)n_in; (void)out_size; (void)ws_size;

    float* ws   = (float*)d_ws;
    float* uhat = ws;                                     // 47,185,920 f (188.7 MB)
    float* xT   = uhat + (size_t)C_CH * UHAT_PER_C;       //  2,359,296 f
    float* v    = xT + (size_t)C_CH * BI;                 //     40,960 f
    float* bij  = v + UHAT_PER_C;                         //     11,520 f
    float* cij  = bij + BU;                               //     11,520 f

    caps_transpose<<<dim3(C_CH / 32, BI / 32), dim3(32, 8), 0, stream>>>(x, xT);
    caps_uhat<<<C_CH, 320, 0, stream>>>(xT, W, uhat, bij, cij);

    for (int it = 0; it < 3; ++it) {
        float* vdst = (it == 2) ? (float*)d_out : v;      // output layout == [B,U,J]
        caps_route_sv<<<UHAT_PER_C / 256, 256, 0, stream>>>(uhat, cij, vdst);
        if (it < 2) caps_agree<<<C_CH, 320, 0, stream>>>(uhat, v, bij, cij);
    }
}